// CMPNNConv_74088185856509
// MI455X (gfx1250) — compile-verified
//
#include <hip/hip_runtime.h>
#include <hip/hip_bf16.h>

typedef float v2f __attribute__((ext_vector_type(2)));
typedef float v8f __attribute__((ext_vector_type(8)));

#define NODE_F 64
#define OUT_F  128
#define CAT_F  192          // 64 + 64 + 64
#define XS_STRIDE 196       // padded LDS row stride (192 % 64banks == 0 -> pad)

// ---------------- zero workspace (deg + h1 + h2, contiguous) ----------------
__global__ void cmpnn_zero(float* __restrict__ p, size_t nfloat) {
    size_t i4 = (size_t)blockIdx.x * blockDim.x + threadIdx.x;
    size_t n4 = nfloat >> 2;
    if (i4 < n4) {
        ((float4*)p)[i4] = make_float4(0.f, 0.f, 0.f, 0.f);
    }
    // tail (nfloat is a multiple of 4 for our sizes, but stay safe)
    if (i4 == 0) {
        for (size_t t = n4 << 2; t < nfloat; ++t) p[t] = 0.f;
    }
}

// ---------------- degree: deg[dst[e]] += 1 ----------------
__global__ void cmpnn_deg(const int* __restrict__ dst, float* __restrict__ deg, int n_edges) {
    int e = blockIdx.x * blockDim.x + threadIdx.x;
    if (e < n_edges) atomicAdd(&deg[dst[e]], 1.0f);
}

// ---------------- norm[n] = max(deg,1)^-0.5 ----------------
__global__ void cmpnn_norm(const float* __restrict__ deg, float* __restrict__ nrm, int n_nodes) {
    int n = blockIdx.x * blockDim.x + threadIdx.x;
    if (n < n_nodes) nrm[n] = 1.0f / sqrtf(fmaxf(deg[n], 1.0f));
}

// ---------------- h1[dst[e]][f] += edge_feat[e][f] ----------------
__global__ void cmpnn_scatter1(const float* __restrict__ edge_feat, const int* __restrict__ dst,
                               float* __restrict__ h1, int n_edges) {
    int idx = blockIdx.x * blockDim.x + threadIdx.x;   // (edge, feat) pair
    if (idx < n_edges * NODE_F) {
        int e = idx >> 6;
        int f = idx & 63;
        atomicAdd(&h1[(size_t)dst[e] * NODE_F + f], edge_feat[idx]);
    }
}

// ---------------- h2[dst[e]][f] += h1[src[e]][f]  (fused gather+scatter) ----------------
__global__ void cmpnn_scatter2(const float* __restrict__ h1, const int* __restrict__ src,
                               const int* __restrict__ dst, float* __restrict__ h2, int n_edges) {
    int idx = blockIdx.x * blockDim.x + threadIdx.x;
    if (idx < n_edges * NODE_F) {
        int e = idx >> 6;
        int f = idx & 63;
        float v = h1[(size_t)src[e] * NODE_F + f];
        atomicAdd(&h2[(size_t)dst[e] * NODE_F + f], v);
    }
}

// ---------------- rst = [node_feat | h1*norm | h2*norm] @ W^T + b  via WMMA f32 ----------------
// Block = 256 threads = 8 waves. Block handles 16 node rows; wave w handles output cols [16w,16w+16).
__global__ __launch_bounds__(256) void cmpnn_gemm(
    const float* __restrict__ node_feat, const float* __restrict__ h1,
    const float* __restrict__ h2, const float* __restrict__ nrm,
    const float* __restrict__ W, const float* __restrict__ bias,
    float* __restrict__ rst, int n_nodes)
{
    __shared__ float Xs[16 * XS_STRIDE];   // 16 rows x 192 cols (padded)

    const int node0 = blockIdx.x * 16;
    const int tid = threadIdx.x;

    // Cooperatively stage the concatenated 16x192 feature tile into LDS.
    for (int i = tid; i < 16 * CAT_F; i += 256) {
        int row = i / CAT_F;
        int col = i - row * CAT_F;
        int node = node0 + row;
        if (node >= n_nodes) node = n_nodes - 1;   // safe clamp (no tail for 50000)
        float v;
        if (col < 64)        v = node_feat[(size_t)node * NODE_F + col];
        else if (col < 128)  v = h1[(size_t)node * NODE_F + (col - 64)]  * nrm[node];
        else                 v = h2[(size_t)node * NODE_F + (col - 128)] * nrm[node];
        Xs[row * XS_STRIDE + col] = v;
    }
    __syncthreads();

    const int wave = tid >> 5;            // 0..7 -> output column tile
    const int lane = tid & 31;
    const int m    = lane & 15;           // A row / D col index helper
    const int kk   = (lane >> 4) << 1;    // lanes 0-15: K=0,1 ; lanes 16-31: K=2,3
    const int col  = wave * 16 + m;       // this lane's B/D column (0..127)

    const float* wrow = W + (size_t)col * CAT_F;  // W[col][k]  (B[k][col] = W[col][k])

    v8f acc = {};
#pragma unroll
    for (int k = 0; k < CAT_F; k += 4) {
        // A fragment: X[m][k+kk], X[m][k+kk+1]   (8B aligned: stride 196*m + even)
        v2f a = *(const v2f*)(&Xs[m * XS_STRIDE + k + kk]);
        // B fragment: W[col][k+kk], W[col][k+kk+1]
        v2f b = *(const v2f*)(wrow + k + kk);
        acc = __builtin_amdgcn_wmma_f32_16x16x4_f32(
            /*neg_a=*/false, a, /*neg_b=*/false, b,
            /*c_mod=*/(short)0, acc, /*reuse_a=*/false, /*reuse_b=*/false);
    }

    const float bv = bias[col];
    const int rbase = (lane >> 4) * 8;    // D: vgpr v -> row v (lanes 0-15) or v+8 (lanes 16-31)
#pragma unroll
    for (int v = 0; v < 8; ++v) {
        int row = node0 + rbase + v;
        if (row < n_nodes)
            rst[(size_t)row * OUT_F + col] = acc[v] + bv;
    }
}

// ---------------- est[e] = rst[src[e]]  (128 floats = 32 lanes x float4) ----------------
__global__ void cmpnn_est(const float* __restrict__ rst, const int* __restrict__ src,
                          float* __restrict__ est, int n_edges) {
    size_t idx = (size_t)blockIdx.x * blockDim.x + threadIdx.x;
    if (idx < (size_t)n_edges * 32) {
        size_t e = idx >> 5;
        int q = (int)(idx & 31);
        int s = src[e];
        float4 v = ((const float4*)(rst + (size_t)s * OUT_F))[q];
        ((float4*)(est + e * OUT_F))[q] = v;
    }
}

extern "C" void kernel_launch(void* const* d_in, const int* in_sizes, int n_in,
                              void* d_out, int out_size, void* d_ws, size_t ws_size,
                              hipStream_t stream) {
    const float* node_feat = (const float*)d_in[0];
    const float* edge_feat = (const float*)d_in[1];
    const float* W         = (const float*)d_in[2];
    const float* bias      = (const float*)d_in[3];
    const int*   src       = (const int*)d_in[4];
    const int*   dst       = (const int*)d_in[5];

    const int n_nodes = in_sizes[0] / NODE_F;   // 50000
    const int n_edges = in_sizes[4];            // 1,000,000

    float* rst = (float*)d_out;                           // (n_nodes, 128)
    float* est = rst + (size_t)n_nodes * OUT_F;           // (n_edges, 128)

    // workspace layout: [deg | h1 | h2 | norm]
    float* ws  = (float*)d_ws;
    float* deg = ws;
    float* h1  = deg + n_nodes;
    float* h2  = h1 + (size_t)n_nodes * NODE_F;
    float* nrm = h2 + (size_t)n_nodes * NODE_F;

    const size_t zero_floats = (size_t)n_nodes * (1 + NODE_F + NODE_F);  // deg+h1+h2
    {
        size_t n4 = (zero_floats + 3) >> 2;
        cmpnn_zero<<<(unsigned)((n4 + 255) / 256), 256, 0, stream>>>(ws, zero_floats);
    }
    cmpnn_deg<<<(n_edges + 255) / 256, 256, 0, stream>>>(dst, deg, n_edges);
    cmpnn_norm<<<(n_nodes + 255) / 256, 256, 0, stream>>>(deg, nrm, n_nodes);

    {
        size_t work = (size_t)n_edges * NODE_F;
        cmpnn_scatter1<<<(unsigned)((work + 255) / 256), 256, 0, stream>>>(edge_feat, dst, h1, n_edges);
        cmpnn_scatter2<<<(unsigned)((work + 255) / 256), 256, 0, stream>>>(h1, src, dst, h2, n_edges);
    }

    cmpnn_gemm<<<(n_nodes + 15) / 16, 256, 0, stream>>>(node_feat, h1, h2, nrm, W, bias, rst, n_nodes);

    {
        size_t work = (size_t)n_edges * 32;
        cmpnn_est<<<(unsigned)((work + 255) / 256), 256, 0, stream>>>(rst, src, est, n_edges);
    }
}